// EncodingLayer_66537633349745
// MI455X (gfx1250) — compile-verified
//
#include <hip/hip_runtime.h>
#include <stdint.h>

// ---------------------------------------------------------------------------
// Fused EncodingLayer + IafPscDelta LIF scan for MI455X (gfx1250, wave32).
//
//   i[b,s,h]  = sum_i x[b,s,i,h] * enc[i,h]          (per-h reduction, K=16)
//   repeat each s for T=10 LIF steps:
//       v = d*v + (1-d)*i - z ; z = (v > 1) && (z == 0)   [r == z_prev]
//   out[b, s*10+t, h] = z ; then zf, vf, rf(==zf) appended.
//
// Memory-bound (~0.43 GB streamed @ 23.3 TB/s => ~18us floor). No GEMM
// structure exists (B-matrix varies along N), so no WMMA; CDNA5 features
// used: async global->LDS (ASYNCcnt), s_wait_asynccnt, global_prefetch_b8,
// non-temporal load/store temporal hints.
// ---------------------------------------------------------------------------

namespace {
constexpr int kB = 32;
constexpr int kS = 64;
constexpr int kI = 16;
constexpr int kH = 2048;
constexpr int kT = 10;
constexpr unsigned kOutSpikes = (unsigned)kB * kS * kT * kH;   // 41,943,040
constexpr float kDecay = 0.951229424500714f;                   // exp(-1/20)
constexpr float kOneMinusDecay = 1.0f - 0.951229424500714f;
}  // namespace

__global__ __launch_bounds__(256) void lif_encoding_fused(
    const float* __restrict__ x,      // [B,S,I,H]
    const float* __restrict__ enc,    // [I,H]
    float* __restrict__ out)          // [B*S*T*H] ++ zf ++ vf ++ rf
{
    __shared__ __align__(16) float s_enc[kI * 256];   // 16 KB tile

    const unsigned tid = threadIdx.x;            // 0..255  -> h within tile
    const unsigned b   = blockIdx.x >> 3;        // 8 blocks per batch row
    const unsigned h0  = (blockIdx.x & 7u) * 256u;
    const unsigned h   = h0 + tid;

    // ---- Stage encoding tile (16 x 256 f32) into LDS with CDNA5 async DMA.
    {
        const unsigned lds_base = (unsigned)(uintptr_t)(&s_enc[0]);
#pragma unroll
        for (int k = 0; k < 4; ++k) {
            const unsigned m    = tid + (unsigned)k * 256u;  // 16B chunk id
            const unsigned fidx = m * 4u;                    // flat f32 idx
            const unsigned row  = fidx >> 8;                 // i
            const unsigned col  = fidx & 255u;               // h offset
            const float* g = enc + (size_t)row * kH + h0 + col;
            const unsigned lds_addr = lds_base + m * 16u;
            asm volatile("global_load_async_to_lds_b128 %0, %1, off"
                         :
                         : "v"(lds_addr), "v"(g)
                         : "memory");
        }
        asm volatile("s_wait_asynccnt 0" ::: "memory");
        __syncthreads();
    }

    // Encoding column for this neuron -> registers (reused for all 64 s).
    float e[kI];
#pragma unroll
    for (int i = 0; i < kI; ++i) e[i] = s_enc[i * 256 + tid];

    float v = 0.0f;
    float z = 0.0f;

    const float* xb = x + ((size_t)b * kS * kI) * kH + h;
    float* ob       = out + ((size_t)b * kS * kT) * kH + h;

    for (int s = 0; s < kS; ++s) {
        const float* xr = xb + (size_t)s * kI * kH;

        // Weighted reduction over I=16 (streamed once -> non-temporal).
        float inp = 0.0f;
#pragma unroll
        for (int i = 0; i < kI; ++i) {
            const float xv = __builtin_nontemporal_load(xr + (size_t)i * kH);
            inp = fmaf(xv, e[i], inp);
        }

        // Prefetch next sequence position's 16 rows (in-bounds only: WGP
        // scope prefetch is non-speculative).
        if (s < kS - 1) {
#pragma unroll
            for (int i = 0; i < kI; ++i)
                __builtin_prefetch(xr + (size_t)(i + kI) * kH, 0, 3);
        }

        // 10 LIF steps driven by the same input current.
        float* o = ob + (size_t)s * kT * kH;
#pragma unroll
        for (int t = 0; t < kT; ++t) {
            v = kDecay * v + kOneMinusDecay * inp - z;   // soft reset by z_prev
            z = (v > 1.0f && z == 0.0f) ? 1.0f : 0.0f;   // spike & refractory
            __builtin_nontemporal_store(z, o + (size_t)t * kH);
        }
    }

    // Final states: (zf, vf, rf) with rf == zf.
    const unsigned n = b * (unsigned)kH + h;
    float* st = out + kOutSpikes;
    st[n] = z;
    st[(unsigned)kB * kH + n] = v;
    st[2u * (unsigned)kB * kH + n] = z;
}

extern "C" void kernel_launch(void* const* d_in, const int* in_sizes, int n_in,
                              void* d_out, int out_size, void* d_ws, size_t ws_size,
                              hipStream_t stream) {
    (void)in_sizes; (void)n_in; (void)out_size; (void)d_ws; (void)ws_size;
    const float* x   = (const float*)d_in[0];
    const float* enc = (const float*)d_in[1];
    float* out       = (float*)d_out;

    dim3 block(256);
    dim3 grid(kB * (kH / 256));   // 256 blocks, 1 thread per (b,h) neuron
    hipLaunchKernelGGL(lif_encoding_fused, grid, block, 0, stream, x, enc, out);
}